// LMHTNeuronPN_6837587935406
// MI455X (gfx1250) — compile-verified
//
#include <hip/hip_runtime.h>
#include <hip/hip_bf16.h>

// LMHT multi-level PN neuron scan, T=8.
//   v += x[t]; out = trunc(clamp(v/th, -4, 4)) * th; v -= out
// Pure streaming workload: 134 MB in + 134 MB out, recurrence held in VGPRs.
// Strategy: 1 thread = 1 float4 column across all 8 timesteps.
//   - all 8 GLOBAL_LOAD_B128 issued before the compute chain (loads are
//     independent of the recurrence), retired one-by-one via s_wait_loadcnt
//   - non-temporal (TH_NT) load/store: zero-reuse stream, skip L2 pollution

typedef float v4f __attribute__((ext_vector_type(4)));

#define T_STEPS 8

__device__ __forceinline__ float pn_quant(float v, float rth, float th) {
    float k = v * rth;
    k = fminf(4.0f, fmaxf(-4.0f, k));   // clamp to [-L, L], L = 4
    k = truncf(k);                       // floor(+)/ceil(-)/0  ==  trunc
    return k * th;
}

__global__ __launch_bounds__(256) void lmht_pn_scan_kernel(
    const float* __restrict__ x,      // [T, B*N] flattened
    const float* __restrict__ vth,    // [1]
    float* __restrict__ out,          // [T, B*N]
    int n_per_t)                      // B*N (multiple of 4)
{
    const int i = (blockIdx.x * 256 + threadIdx.x) * 4;
    if (i >= n_per_t) return;

    const float th  = vth[0];
    const float rth = 1.0f / th;      // hoisted: one divide per thread

    // Issue all T loads up front: 8 outstanding B128 loads per thread.
    v4f xs[T_STEPS];
    const float* px = x + i;
#pragma unroll
    for (int t = 0; t < T_STEPS; ++t) {
        xs[t] = __builtin_nontemporal_load(
            (const v4f*)(px + (size_t)t * (size_t)n_per_t));
    }

    // Recurrence lives entirely in registers.
    v4f v = (v4f)0.0f;
    float* po = out + i;
#pragma unroll
    for (int t = 0; t < T_STEPS; ++t) {
        v += xs[t];
        v4f o;
        o.x = pn_quant(v.x, rth, th);
        o.y = pn_quant(v.y, rth, th);
        o.z = pn_quant(v.z, rth, th);
        o.w = pn_quant(v.w, rth, th);
        v -= o;
        __builtin_nontemporal_store(
            o, (v4f*)(po + (size_t)t * (size_t)n_per_t));
    }
}

extern "C" void kernel_launch(void* const* d_in, const int* in_sizes, int n_in,
                              void* d_out, int out_size, void* d_ws, size_t ws_size,
                              hipStream_t stream) {
    const float* x   = (const float*)d_in[0];   // [T, B, N] float32
    const float* vth = (const float*)d_in[1];   // [1] float32
    float* out       = (float*)d_out;           // [T, B, N] float32

    const int n_total = in_sizes[0];            // T * B * N
    const int n_per_t = n_total / T_STEPS;      // B * N (4,194,304)

    const int nvec   = (n_per_t + 3) / 4;       // float4 work-items
    const int block  = 256;                     // 8 waves (wave32) per block
    const int grid   = (nvec + block - 1) / block;

    lmht_pn_scan_kernel<<<grid, block, 0, stream>>>(x, vth, out, n_per_t);
}